// SpatialAttention_82025285419552
// MI455X (gfx1250) — compile-verified
//
#include <hip/hip_runtime.h>
#include <hip/hip_bf16.h>

// ---------------- constants (match reference) ----------------
#define IN_DIM 128
#define HC     256    // heads * channels
#define HEADS  4
#define CH     64

typedef __attribute__((ext_vector_type(2))) float v2f;
typedef __attribute__((ext_vector_type(8))) float v8f;
typedef __attribute__((ext_vector_type(4))) unsigned int v4u;
typedef __attribute__((ext_vector_type(8))) int v8i;
typedef __attribute__((ext_vector_type(4))) int v4i;

// ---------------- TDM: stage a K x 16 column panel of row-major B[K,N] into LDS ---------
// D# per CDNA5 ISA: 2-D tensor, data_size=4B,
// tensor_dim0 = N (row length), tensor_dim0_stride = N, tile = 16 x K,
// global_addr = panel start, lds dest = compact row-major 16-wide tile.
__device__ __forceinline__ void tdm_load_panel(const float* gptr, unsigned lds_byte_off,
                                               int n_total, int k_rows) {
  unsigned long long ga = (unsigned long long)(uintptr_t)gptr;
  v4u g0;
  g0.x = 1u;                                            // count=1, user mode
  g0.y = lds_byte_off;                                  // lds_addr
  g0.z = (unsigned)(ga & 0xFFFFFFFFu);                  // global_addr[31:0]
  g0.w = (unsigned)((ga >> 32) & 0x01FFFFFFu)           // global_addr[56:32]
         | 0x80000000u;                                 // type=2 ("image")
  v8i g1;
  g1[0] = 0x00020000;                                   // data_size=2 (4 bytes)
  g1[1] = (n_total & 0xFFFF) << 16;                     // tensor_dim0[15:0]
  g1[2] = ((unsigned)n_total >> 16) | ((k_rows & 0xFFFF) << 16); // dim0 hi | dim1 lo
  g1[3] = ((unsigned)k_rows >> 16) | (16 << 16);        // dim1 hi | tile_dim0=16
  g1[4] = k_rows & 0xFFFF;                              // tile_dim1 = K rows (tile_dim2=0)
  g1[5] = n_total;                                      // tensor_dim0_stride[31:0]
  g1[6] = 0;                                            // stride hi | dim1_stride lo
  g1[7] = 0;
  v4i z4 = {0, 0, 0, 0};
  v8i z8 = {0, 0, 0, 0, 0, 0, 0, 0};
  __builtin_amdgcn_tensor_load_to_lds(g0, g1, z4, z4, z8, 0);
}

// ---------------- fused projection GEMM -------------------------------------------------
// One block = 8 waves = 8 consecutive M tiles at one N tile.  Each wave computes the
// 16x16 tile of ALL FOUR outputs (q,k,v,skip) so A is read once per K-step.
// B panels (4 x K x 16 = 32 KB) staged into LDS by the TDM.
__global__ __launch_bounds__(256) void proj_fused_wmma(
    const float* __restrict__ A,
    const float* __restrict__ Bq, const float* __restrict__ Bk,
    const float* __restrict__ Bv, const float* __restrict__ Bs,
    const float* __restrict__ bq, const float* __restrict__ bk,
    const float* __restrict__ bv, const float* __restrict__ bs,
    float* __restrict__ Cq, float* __restrict__ Ck,
    float* __restrict__ Cv, float* __restrict__ Cs,
    int M, int K, int N) {
  __shared__ float lds[4 * IN_DIM * 16];                // 4 panels of K(=128) x 16

  const int nT    = N >> 4;
  const int tileN = blockIdx.x % nT;
  const int mBlk  = blockIdx.x / nT;
  const int wid   = threadIdx.x >> 5;
  const int lane  = threadIdx.x & 31;
  const int tileM = mBlk * 8 + wid;

  // wave 0 issues the 4 tensor loads (descriptor fields are block-uniform -> SGPRs)
  if (threadIdx.x < 32) {
    const unsigned base = (unsigned)(uintptr_t)lds;
    const unsigned panelB = (unsigned)(K * 16 * 4);     // bytes per panel
    tdm_load_panel(Bq + tileN * 16, base + 0 * panelB, N, K);
    tdm_load_panel(Bk + tileN * 16, base + 1 * panelB, N, K);
    tdm_load_panel(Bv + tileN * 16, base + 2 * panelB, N, K);
    tdm_load_panel(Bs + tileN * 16, base + 3 * panelB, N, K);
    __builtin_amdgcn_s_wait_tensorcnt(0);
  }
  __syncthreads();

  const int half = lane >> 4;          // selects K pair (0: K0,K1 ; 1: K2,K3)
  const int l15  = lane & 15;
  const int row  = tileM * 16 + l15;
  const int col  = tileN * 16 + l15;

  v8f aq, ak, av, as_;
  {
    const float q0 = bq[col], k0 = bk[col], v0 = bv[col], s0 = bs[col];
    #pragma unroll
    for (int j = 0; j < 8; ++j) { aq[j] = q0; ak[j] = k0; av[j] = v0; as_[j] = s0; }
  }

  const float* __restrict__ Arow = A + (size_t)row * K + half * 2;
  const int P = K * 16;                                  // floats per LDS panel

  #pragma unroll 4
  for (int k0 = 0; k0 < K; k0 += 4) {
    v2f a = *(const v2f*)(Arow + k0);                    // A[row][k0+2h .. +1]
    const int kk = (k0 + 2 * half) * 16 + l15;           // panel index of B[k][col]
    v2f b;
    b.x = lds[0 * P + kk];      b.y = lds[0 * P + kk + 16];
    aq  = __builtin_amdgcn_wmma_f32_16x16x4_f32(false, a, false, b, (short)0, aq,  false, false);
    b.x = lds[1 * P + kk];      b.y = lds[1 * P + kk + 16];
    ak  = __builtin_amdgcn_wmma_f32_16x16x4_f32(false, a, false, b, (short)0, ak,  false, false);
    b.x = lds[2 * P + kk];      b.y = lds[2 * P + kk + 16];
    av  = __builtin_amdgcn_wmma_f32_16x16x4_f32(false, a, false, b, (short)0, av,  false, false);
    b.x = lds[3 * P + kk];      b.y = lds[3 * P + kk + 16];
    as_ = __builtin_amdgcn_wmma_f32_16x16x4_f32(false, a, false, b, (short)0, as_, false, false);
  }

  const int baseRow = tileM * 16 + 8 * half;
  #pragma unroll
  for (int j = 0; j < 8; ++j) {
    const size_t o = (size_t)(baseRow + j) * N + col;
    Cq[o] = aq[j];  Ck[o] = ak[j];  Cv[o] = av[j];  Cs[o] = as_[j];
  }
}

// ---------------- single-matrix GEMM (output projection), B panel staged via TDM --------
__global__ __launch_bounds__(256) void gemm_wmma_lds(
    const float* __restrict__ A, const float* __restrict__ B,
    const float* __restrict__ bias, float* __restrict__ C,
    int M, int K, int N) {
  __shared__ float lds[HC * 16];                         // K(<=256) x 16 panel

  const int nT    = N >> 4;
  const int tileN = blockIdx.x % nT;
  const int mBlk  = blockIdx.x / nT;
  const int wid   = threadIdx.x >> 5;
  const int lane  = threadIdx.x & 31;
  const int tileM = mBlk * 8 + wid;

  if (threadIdx.x < 32) {
    tdm_load_panel(B + tileN * 16, (unsigned)(uintptr_t)lds, N, K);
    __builtin_amdgcn_s_wait_tensorcnt(0);
  }
  __syncthreads();

  const int half = lane >> 4;
  const int l15  = lane & 15;
  const int row  = tileM * 16 + l15;
  const int col  = tileN * 16 + l15;

  v8f acc;
  const float b0 = bias[col];
  #pragma unroll
  for (int j = 0; j < 8; ++j) acc[j] = b0;

  const float* __restrict__ Arow = A + (size_t)row * K + half * 2;

  #pragma unroll 4
  for (int k0 = 0; k0 < K; k0 += 4) {
    v2f a = *(const v2f*)(Arow + k0);
    const int kk = (k0 + 2 * half) * 16 + l15;
    v2f b;
    b.x = lds[kk];
    b.y = lds[kk + 16];
    acc = __builtin_amdgcn_wmma_f32_16x16x4_f32(false, a, false, b, (short)0, acc, false, false);
  }

  const int baseRow = tileM * 16 + 8 * half;
  #pragma unroll
  for (int j = 0; j < 8; ++j)
    C[(size_t)(baseRow + j) * N + col] = acc[j];
}

// ---------------- monotonic float<->uint encoding for atomic max ----------------
__device__ __forceinline__ unsigned f2ord(float f) {
  unsigned u = __float_as_uint(f);
  return u ^ (unsigned)(((int)u >> 31) | 0x80000000);
}
__device__ __forceinline__ float ord2f(unsigned k) {
  unsigned u = (k & 0x80000000u) ? (k ^ 0x80000000u) : ~k;
  return __uint_as_float(u);
}

// e_proj channel c of one edge: edge_attr[e,:3] @ We[:, c]
__device__ __forceinline__ float edge_proj(const float* __restrict__ We,
                                           float ea0, float ea1, float ea2, int c) {
  return ea0 * We[c] + ea1 * We[HC + c] + ea2 * We[2 * HC + c];
}

// ---------------- init segment-max / denom ----------------
__global__ void init_seg(unsigned* __restrict__ mmax, float* __restrict__ den, int n) {
  int t = blockIdx.x * blockDim.x + threadIdx.x;
  if (t < n) { mmax[t] = 0u; den[t] = 0.0f; }            // key 0 encodes below -FLT_MAX
}

// ---------------- per-edge attention logits + segment max (1 wave / edge) ----------------
__global__ __launch_bounds__(256) void edge_logits(
    const float* __restrict__ q, const float* __restrict__ k,
    const float* __restrict__ edge_attr, const float* __restrict__ We,
    const int* __restrict__ edge_index, float* __restrict__ alpha,
    unsigned* __restrict__ mmax, int E) {
  const int w = (blockIdx.x * blockDim.x + threadIdx.x) >> 5;
  if (w >= E) return;
  const int lane = threadIdx.x & 31;
  const int src = edge_index[w];
  const int dst = edge_index[E + w];

  const float ea0 = edge_attr[w * 3 + 0];
  const float ea1 = edge_attr[w * 3 + 1];
  const float ea2 = edge_attr[w * 3 + 2];

  const int c0 = lane * 8;                               // head = lane>>3
  const float* __restrict__ qr = q + (size_t)dst * HC + c0;
  const float* __restrict__ kr = k + (size_t)src * HC + c0;

  float s = 0.0f;
  #pragma unroll
  for (int j = 0; j < 8; ++j) {
    float e = edge_proj(We, ea0, ea1, ea2, c0 + j);
    s += qr[j] * (kr[j] + e);
  }
  s += __shfl_xor(s, 1);
  s += __shfl_xor(s, 2);
  s += __shfl_xor(s, 4);
  s *= 0.125f;                                           // 1/sqrt(C), C=64

  if ((lane & 7) == 0) {
    const int h = lane >> 3;
    alpha[(size_t)w * HEADS + h] = s;
    atomicMax(&mmax[dst * HEADS + h], f2ord(s));
  }
}

// ---------------- exp(alpha - max) + segment sum (1 thread / edge-head) ----------------
__global__ void edge_exp(const int* __restrict__ edge_index,
                         const unsigned* __restrict__ mmax,
                         float* __restrict__ alpha, float* __restrict__ den, int E) {
  int t = blockIdx.x * blockDim.x + threadIdx.x;
  if (t >= E * HEADS) return;
  const int w = t >> 2;
  const int h = t & 3;
  const int dst = edge_index[E + w];
  const float m = ord2f(mmax[dst * HEADS + h]);
  const float ex = expf(alpha[t] - m);
  alpha[t] = ex;
  atomicAdd(&den[dst * HEADS + h], ex);
}

// ---------------- weighted message scatter (1 wave / edge) ----------------
__global__ __launch_bounds__(256) void edge_scatter(
    const float* __restrict__ v, const float* __restrict__ edge_attr,
    const float* __restrict__ We, const int* __restrict__ edge_index,
    const float* __restrict__ alpha, const float* __restrict__ den,
    float* __restrict__ agg, int E) {
  const int w = (blockIdx.x * blockDim.x + threadIdx.x) >> 5;
  if (w >= E) return;
  const int lane = threadIdx.x & 31;
  const int src = edge_index[w];
  const int dst = edge_index[E + w];
  const int h = lane >> 3;

  const float ex = alpha[(size_t)w * HEADS + h];
  const float d  = den[dst * HEADS + h];
  const float a  = ex / (d + 1e-16f);

  const float ea0 = edge_attr[w * 3 + 0];
  const float ea1 = edge_attr[w * 3 + 1];
  const float ea2 = edge_attr[w * 3 + 2];

  const int c0 = lane * 8;
  const float* __restrict__ vr = v + (size_t)src * HC + c0;
  float* __restrict__ ar = agg + (size_t)dst * HC + c0;
  #pragma unroll
  for (int j = 0; j < 8; ++j) {
    float e = edge_proj(We, ea0, ea1, ea2, c0 + j);
    atomicAdd(&ar[j], (vr[j] + e) * a);
  }
}

// ---------------- host-side launch ----------------
extern "C" void kernel_launch(void* const* d_in, const int* in_sizes, int n_in,
                              void* d_out, int out_size, void* d_ws, size_t ws_size,
                              hipStream_t stream) {
  const float* x     = (const float*)d_in[0];
  const int*   eidx  = (const int*)d_in[1];              // [2, E]
  const float* eattr = (const float*)d_in[2];
  const float* Wq    = (const float*)d_in[3];
  const float* bq    = (const float*)d_in[4];
  const float* Wk    = (const float*)d_in[5];
  const float* bk    = (const float*)d_in[6];
  const float* Wv    = (const float*)d_in[7];
  const float* bv    = (const float*)d_in[8];
  const float* We    = (const float*)d_in[9];
  const float* Wskip = (const float*)d_in[10];
  const float* bskip = (const float*)d_in[11];
  const float* Wout  = (const float*)d_in[12];
  const float* bout  = (const float*)d_in[13];
  float*       out   = (float*)d_out;

  const int N = in_sizes[0] / IN_DIM;                    // 32000
  const int E = in_sizes[1] / 2;                         // 256000

  // workspace carve-up
  float* ws    = (float*)d_ws;
  float* q     = ws;
  float* k     = q   + (size_t)N * HC;
  float* v     = k   + (size_t)N * HC;
  float* agg   = v   + (size_t)N * HC;                   // initialized by skip GEMM
  float* alpha = agg + (size_t)N * HC;                   // [E, H]
  unsigned* mmax = (unsigned*)(alpha + (size_t)E * HEADS);
  float* den   = (float*)(mmax + (size_t)N * HEADS);

  const int BLK = 256;

  // 1) fused projections: q,k,v,agg=skip  (M=N, K=128, N=256); 8 M-tiles per block
  {
    int blocks = (HC / 16) * (N / 128);                  // 16 * 250
    proj_fused_wmma<<<blocks, BLK, 0, stream>>>(x, Wq, Wk, Wv, Wskip,
                                                bq, bk, bv, bskip,
                                                q, k, v, agg, N, IN_DIM, HC);
  }

  // 2) init segment max / denom
  {
    int n = N * HEADS;
    init_seg<<<(n + BLK - 1) / BLK, BLK, 0, stream>>>(mmax, den, n);
  }

  // 3) per-edge logits + segment max
  edge_logits<<<(E * 32 + BLK - 1) / BLK, BLK, 0, stream>>>(q, k, eattr, We, eidx,
                                                            alpha, mmax, E);

  // 4) exp + denominator
  edge_exp<<<(E * HEADS + BLK - 1) / BLK, BLK, 0, stream>>>(eidx, mmax, alpha, den, E);

  // 5) weighted message scatter into agg (holds skip term)
  edge_scatter<<<(E * 32 + BLK - 1) / BLK, BLK, 0, stream>>>(v, eattr, We, eidx,
                                                             alpha, den, agg, E);

  // 6) output projection: out = agg @ Wout + bout  (M=N, K=256, N=128)
  {
    int blocks = (IN_DIM / 16) * (N / 128);              // 8 * 250
    gemm_wmma_lds<<<blocks, BLK, 0, stream>>>(agg, Wout, bout, out, N, HC, IN_DIM);
  }
}